// DifferentiableHeadRenderer_28217935135055
// MI455X (gfx1250) — compile-verified
//
#include <hip/hip_runtime.h>
#include <stdint.h>

#define N_GAUSS   2048
#define HEIGHT    192
#define WIDTH     192
#define HW        (HEIGHT * WIDTH)
#define ALPHA_MAX 0.999f
#define ALPHA_MIN (1.0f / 255.0f)

#define REC    12                 // floats per sorted gaussian record (48 B = 3 x b128)
#define CHUNK  512                // gaussians per LDS chunk
#define NCHUNK (N_GAUSS / CHUNK)  // 4
#define TPB    256                // 8 wave32 per block
#define T_EPS  1e-6f
#define SUB    8                  // gaussians per branch-free sub-block

// ---- CDNA5 async global->LDS path (guarded; fallback = sync copy) ----------
#if __has_builtin(__builtin_amdgcn_global_load_async_to_lds_b128)
#define HAVE_ASYNC 1
#else
#define HAVE_ASYNC 0
#endif

typedef int v4i __attribute__((ext_vector_type(4)));

__device__ __forceinline__ void async_copy_b128(const float4* __restrict__ g,
                                                float4* __restrict__ l) {
#if HAVE_ASYNC
  // (global src [AS1, prints as __device__], LDS dst [AS3], imm offset, cpol)
  __builtin_amdgcn_global_load_async_to_lds_b128((v4i*)g, (v4i*)l,
                                                 /*imm offset*/ 0,
                                                 /*cpol*/ 0);
#else
  *l = *g;
#endif
}

template <int N>
__device__ __forceinline__ void wait_asynccnt() {
#if HAVE_ASYNC
#if __has_builtin(__builtin_amdgcn_s_wait_asynccnt)
  __builtin_amdgcn_s_wait_asynccnt(N);
#else
  asm volatile("s_wait_asynccnt %0" ::"i"(N) : "memory");
#endif
#endif
}

// ---- Kernel 1: O(N^2) stable rank-sort by depth + gather into packed records
__global__ __launch_bounds__(TPB) void gs_sort_gather(
    const float* __restrict__ means2d,   // [N,2]
    const float* __restrict__ conics,    // [N,3]
    const float* __restrict__ colors,    // [N,3]
    const float* __restrict__ opacities, // [N]
    const float* __restrict__ depths,    // [N]
    float* __restrict__ ws) {            // [N*REC]
  __shared__ float sdep[N_GAUSS];
  const int tid = threadIdx.x;
  for (int j = tid; j < N_GAUSS; j += TPB) sdep[j] = depths[j];
  __syncthreads();

  const int i  = blockIdx.x * TPB + tid;   // grid sized exactly to N_GAUSS
  const float di = sdep[i];
  int rank = 0;
#pragma unroll 8
  for (int j = 0; j < N_GAUSS; ++j) {
    const float dj = sdep[j];
    rank += (dj < di) || (dj == di && j < i);   // stable (argsort) tie-break
  }

  float* r = ws + (size_t)rank * REC;
  r[0]  = means2d[2 * i + 0];
  r[1]  = means2d[2 * i + 1];
  r[2]  = conics[3 * i + 0];   // a
  r[3]  = conics[3 * i + 1];   // b
  r[4]  = conics[3 * i + 2];   // c
  r[5]  = opacities[i];
  r[6]  = colors[3 * i + 0];
  r[7]  = colors[3 * i + 1];
  r[8]  = colors[3 * i + 2];
  r[9]  = 0.f; r[10] = 0.f; r[11] = 0.f;
}

// ---- Kernel 2: per-pixel front-to-back compositing, double-buffered LDS ----
__device__ __forceinline__ void issue_chunk(const float4* __restrict__ ws4,
                                            int chunk, float4* __restrict__ dst,
                                            int tid) {
  const float4* src = ws4 + (size_t)chunk * CHUNK * 3;
#pragma unroll
  for (int s = 0; s < CHUNK / TPB; ++s) {
    const int lg = s * TPB + tid;
    async_copy_b128(src + lg * 3 + 0, dst + lg * 3 + 0);
    async_copy_b128(src + lg * 3 + 1, dst + lg * 3 + 1);
    async_copy_b128(src + lg * 3 + 2, dst + lg * 3 + 2);
  }
}

__global__ __launch_bounds__(TPB) void gs_render(
    const float* __restrict__ ws,  // sorted records [N*REC]
    const float* __restrict__ bg,  // [3]
    float* __restrict__ out) {     // [3,H,W]
  __shared__ float4 lds[2][CHUNK * 3];   // 2 x 24 KB

  const int tid = threadIdx.x;
  const int p   = blockIdx.x * TPB + tid;
  const float px = (float)(p % WIDTH) + 0.5f;
  const float py = (float)(p / WIDTH) + 0.5f;

  const float4* ws4 = (const float4*)ws;

  issue_chunk(ws4, 0, lds[0], tid);

  float T = 1.0f, accR = 0.0f, accG = 0.0f, accB = 0.0f;

  for (int k = 0; k < NCHUNK; ++k) {
    if (k + 1 < NCHUNK) {
      issue_chunk(ws4, k + 1, lds[(k + 1) & 1], tid);
      wait_asynccnt<6>();   // chunk-(k+1) (6 ops/wave) may remain in flight
    } else {
      wait_asynccnt<0>();
    }
    __syncthreads();        // all waves' chunk-k data resident in LDS

    const float4* buf = lds[k & 1];
    for (int g0 = 0; g0 < CHUNK; g0 += SUB) {
      if (T < T_EPS) break;               // per-lane early-out, 1 check / 8 gauss
#pragma unroll
      for (int u = 0; u < SUB; ++u) {     // branch-free unrolled body
        const int g = g0 + u;
        const float4 v0 = buf[g * 3 + 0];   // mx, my, a, b   (LDS broadcast)
        const float4 v1 = buf[g * 3 + 1];   // c, opac, r, g
        const float  cb = buf[g * 3 + 2].x; // b (color)

        const float dx = px - v0.x;
        const float dy = py - v0.y;
        const float sigma =
            0.5f * (v0.z * dx * dx + v1.x * dy * dy) + v0.w * dx * dy;
        float al = fminf(ALPHA_MAX, v1.y * __expf(-sigma));
        al = (sigma > 0.0f && al >= ALPHA_MIN) ? al : 0.0f;
        const float w = T * al;
        accR = fmaf(w, v1.z, accR);
        accG = fmaf(w, v1.w, accG);
        accB = fmaf(w, cb, accB);
        T *= (1.0f - al);
      }
    }
    __syncthreads();        // done reading buf before it is overwritten
  }

  out[0 * HW + p] = fmaf(bg[0], T, accR);
  out[1 * HW + p] = fmaf(bg[1], T, accG);
  out[2 * HW + p] = fmaf(bg[2], T, accB);
}

// ---- Host-side launch ------------------------------------------------------
extern "C" void kernel_launch(void* const* d_in, const int* in_sizes, int n_in,
                              void* d_out, int out_size, void* d_ws,
                              size_t ws_size, hipStream_t stream) {
  const float* means2d   = (const float*)d_in[0];
  const float* conics    = (const float*)d_in[1];
  const float* colors    = (const float*)d_in[2];
  const float* opacities = (const float*)d_in[3];
  const float* depths    = (const float*)d_in[4];
  const float* background= (const float*)d_in[5];
  float* out = (float*)d_out;
  float* ws  = (float*)d_ws;   // needs N_GAUSS*REC*4 = 96 KB

  gs_sort_gather<<<N_GAUSS / TPB, TPB, 0, stream>>>(
      means2d, conics, colors, opacities, depths, ws);

  gs_render<<<HW / TPB, TPB, 0, stream>>>(ws, background, out);
}